// FusedLinearRowSum_43800076485439
// MI455X (gfx1250) — compile-verified
//
#include <hip/hip_runtime.h>
#include <hip/hip_bf16.h>

typedef __attribute__((ext_vector_type(2))) float v2f;
typedef __attribute__((ext_vector_type(4))) float v4f;
typedef __attribute__((ext_vector_type(8))) float v8f;
typedef __attribute__((ext_vector_type(4))) int v4i;

#define K_DIM 4096
#define N_ROWS 16384
#define TILE_ROWS 128
#define TILE_K 64
#define PITCH 68  // 64 + 4 pad floats: row stride 272B (16B aligned), kills LDS bank conflicts

#if __has_builtin(__builtin_amdgcn_global_load_async_to_lds_b128)
#define HAVE_ASYNC 1
#else
#define HAVE_ASYNC 0
#endif

typedef __attribute__((address_space(1))) v4i* gptr4;
typedef __attribute__((address_space(3))) v4i* lptr4;

// 16-byte global -> LDS copy. Async DMA path (ASYNCcnt) when available,
// else synchronous VGPR round-trip with a non-temporal load hint.
__device__ __forceinline__ void cp16(const float* __restrict__ g, float* l) {
#if HAVE_ASYNC
    __builtin_amdgcn_global_load_async_to_lds_b128(
        (gptr4)g, (lptr4)l, /*offset=*/0, /*cpol=*/0);
#else
    *reinterpret_cast<v4f*>(l) =
        __builtin_nontemporal_load(reinterpret_cast<const v4f*>(g));
#endif
}

__device__ __forceinline__ void wait_async0() {
#if HAVE_ASYNC
#if __has_builtin(__builtin_amdgcn_s_wait_asynccnt)
    __builtin_amdgcn_s_wait_asynccnt(0);
#else
    asm volatile("s_wait_asynccnt 0x0" ::: "memory");
#endif
#endif
}

// ---------------------------------------------------------------------------
// Kernel 1: w_colsum[k] = sum_j weight[j,k]  (blocks 0..15)
//           ws[4096]    = sum(bias)          (block 16)
// Streaming single-touch reads -> non-temporal hint.
// ---------------------------------------------------------------------------
__global__ __launch_bounds__(256) void colsum_kernel(const float* __restrict__ w,
                                                     const float* __restrict__ bias,
                                                     float* __restrict__ ws) {
    __shared__ float red[256];
    const int tid = threadIdx.x;
    if (blockIdx.x < 16) {
        const int col = blockIdx.x * 256 + tid;
        float s0 = 0.f, s1 = 0.f, s2 = 0.f, s3 = 0.f;
        for (int r = 0; r < K_DIM; r += 4) {
            s0 += __builtin_nontemporal_load(&w[(size_t)(r + 0) * K_DIM + col]);
            s1 += __builtin_nontemporal_load(&w[(size_t)(r + 1) * K_DIM + col]);
            s2 += __builtin_nontemporal_load(&w[(size_t)(r + 2) * K_DIM + col]);
            s3 += __builtin_nontemporal_load(&w[(size_t)(r + 3) * K_DIM + col]);
        }
        ws[col] = (s0 + s1) + (s2 + s3);
    } else {
        float s = 0.f;
        for (int i = tid; i < K_DIM; i += 256) s += bias[i];
        red[tid] = s;
        __syncthreads();
        for (int off = 128; off > 0; off >>= 1) {
            if (tid < off) red[tid] += red[tid + off];
            __syncthreads();
        }
        if (tid == 0) ws[K_DIM] = red[0];
    }
}

// ---------------------------------------------------------------------------
// Kernel 2: out[i] = x[i,:] . w_colsum + bias_sum via V_WMMA_F32_16X16X4_F32.
// 256 threads = 8 wave32; each wave owns 16 rows. Double-buffered LDS tiles
// (2 x 128x64 f32, ~70KB of the 320KB/WGP pool) filled by async global->LDS
// DMA; next tile's loads are in flight while WMMAs consume the current tile.
// ---------------------------------------------------------------------------
__global__ __launch_bounds__(256) void gemv_wmma_kernel(const float* __restrict__ x,
                                                        const float* __restrict__ wsum,
                                                        const float* __restrict__ bsum_p,
                                                        float* __restrict__ out) {
    __shared__ float lx[2][TILE_ROWS * PITCH];  // 2 x 34,816 B
    __shared__ float lw[2][TILE_K];

    const int tid   = threadIdx.x;
    const int lane  = tid & 31;
    const int wave  = tid >> 5;
    const int rBase = blockIdx.x * TILE_ROWS;

    // WMMA A-fragment coords (32-bit 16x4 layout, ISA 7.12.2):
    //   lanes 0-15: K = {0,1}; lanes 16-31: K = {2,3}; M = lane & 15
    const int m       = lane & 15;
    const int koff    = (lane >> 4) * 2;
    const int waveRow = wave * 16;

    // cooperative-load coords: 16 threads x float4 cover one 64-float row
    const int ldRow = tid >> 4;        // 0..15
    const int ldCol = (tid & 15) * 4;  // 0,4,...,60

    auto issue_tile = [&](int buf, int kb) {
        #pragma unroll
        for (int i = 0; i < 8; ++i) {
            const int row = ldRow + i * 16;
            cp16(x + (size_t)(rBase + row) * K_DIM + kb + ldCol,
                 &lx[buf][row * PITCH + ldCol]);
        }
        if (tid < 16) cp16(wsum + kb + tid * 4, &lw[buf][tid * 4]);
    };

    v8f c = {};  // D accumulator (16x16 f32 = 8 VGPRs)

    issue_tile(0, 0);
    int buf = 0;
    for (int kb = 0; kb < K_DIM; kb += TILE_K, buf ^= 1) {
        wait_async0();     // my async writes for tile `buf` have landed in LDS
        __syncthreads();   // everyone's have
        if (kb + TILE_K < K_DIM) issue_tile(buf ^ 1, kb + TILE_K);

        #pragma unroll
        for (int kk = 0; kk < TILE_K; kk += 4) {
            v2f a = *reinterpret_cast<const v2f*>(
                &lx[buf][(waveRow + m) * PITCH + kk + koff]);  // ds_load_b64
            v2f b;
            b.x = lw[buf][kk + koff];      // B row K=koff   (broadcast read)
            b.y = lw[buf][kk + koff + 1];  // B row K=koff+1
            c = __builtin_amdgcn_wmma_f32_16x16x4_f32(
                /*neg_a=*/false, a, /*neg_b=*/false, b,
                /*c_mod=*/(short)0, c, /*reuse_a=*/false, /*reuse_b=*/false);
        }
        __syncthreads();   // all reads of `buf` done before it is refilled
    }

    // D[m,j] identical for all j. C/D layout: VGPR v lane 0 -> row v,
    // VGPR v lane 16 -> row v+8. Lanes 0 and 16 emit the 16 row results.
    const float bs = *bsum_p;
    if (lane == 0) {
        #pragma unroll
        for (int v = 0; v < 8; ++v) out[rBase + waveRow + v] = c[v] + bs;
    } else if (lane == 16) {
        #pragma unroll
        for (int v = 0; v < 8; ++v) out[rBase + waveRow + 8 + v] = c[v] + bs;
    }
}

extern "C" void kernel_launch(void* const* d_in, const int* in_sizes, int n_in,
                              void* d_out, int out_size, void* d_ws, size_t ws_size,
                              hipStream_t stream) {
    const float* x    = (const float*)d_in[0];  // [16384, 4096]
    const float* w    = (const float*)d_in[1];  // [4096, 4096]
    const float* bias = (const float*)d_in[2];  // [4096]
    float* out = (float*)d_out;                 // [16384]
    float* ws  = (float*)d_ws;                  // [4097] f32: w_colsum + bias_sum

    colsum_kernel<<<17, 256, 0, stream>>>(w, bias, ws);
    gemv_wmma_kernel<<<N_ROWS / TILE_ROWS, 256, 0, stream>>>(x, ws, ws + K_DIM, out);
}